// DeltaRule_90795608638291
// MI455X (gfx1250) — compile-verified
//
#include <hip/hip_runtime.h>
#include <hip/hip_bf16.h>

// Linear recurrence per column (T=8192 steps, B=4096 columns):
//   g0 = lr*(1-x_{t-1}); g1 = lr*x_{t-1}
//   p0 += g0*(x_t - p0); p1 += g1*(x_t - p1)
//   pred_t = p0*(1-x_t) + p1*x_t
// Affine in (p0,p1) with input-only coefficients -> chunked parallel scan.
// Phase1: per-chunk affine composition (async global->LDS staging).
// Phase2: tiny serial composition over chunks.
// Phase3: replay chunks from entry states (TDM tensor_load_to_lds staging).

#define BCOLS   4096      // columns (B)
#define CHUNK   128       // time steps per chunk
#define COLS    256       // columns per workgroup (== blockDim.x)
#define STEP    16        // time steps staged per LDS tile
#define NTILES  (CHUNK / STEP)           // 8
#define LOADS_PER_TILE 4                 // async path: b128 loads per thread per tile
#define NWAVES  (COLS / 32)              // 8 waves per block
#define RPW     (STEP / NWAVES)          // 2 tile rows per wave (TDM slab)

#if __has_builtin(__builtin_amdgcn_tensor_load_to_lds)
#define HAVE_TDM 1
#else
#define HAVE_TDM 0
#endif

typedef __attribute__((ext_vector_type(4))) unsigned int v4u;
typedef __attribute__((ext_vector_type(8))) int          v8i;
typedef __attribute__((ext_vector_type(4))) int          v4i;

__device__ __forceinline__ float clamp01(float v) {
    return v < 0.0f ? 0.0f : (v > 1.0f ? 1.0f : v);
}

__device__ __forceinline__ unsigned lds_off(const void* p) {
    // generic pointer to __shared__: low 32 bits are the LDS byte offset,
    // which is what async/TDM LDS address fields expect.
    return (unsigned)(unsigned long long)p;
}

// ---- Async-load staging (per-lane DMA, ASYNCcnt) --------------------------
// One 16-step x 256-col tile (16KB): 1024 x 16B chunks; each of 256 threads
// issues 4 b128 async loads -> ASYNCcnt += 4 per wave per tile.
__device__ __forceinline__ void issue_tile_async(const float* __restrict__ x, int t0,
                                                 int B, int col0, unsigned lbase, int tid) {
    const unsigned long long gbase =
        (unsigned long long)(const void*)(x + (size_t)t0 * B + col0);
#pragma unroll
    for (int i = 0; i < LOADS_PER_TILE; ++i) {
        int f   = tid + i * COLS;        // 0..1023
        int row = f >> 6;                // 64 x 16B chunks per 1KB row
        int cw  = f & 63;
        unsigned long long g = gbase + (unsigned long long)row * (unsigned)(B * 4)
                                     + (unsigned)(cw * 16);
        unsigned l = lbase + (unsigned)(row * COLS * 4 + cw * 16);
        asm volatile("global_load_async_to_lds_b128 %0, %1, off"
                     :: "v"(l), "v"(g) : "memory");
    }
}
__device__ __forceinline__ void wait_async_le4() {
    asm volatile("s_wait_asynccnt 4" ::: "memory");
}
__device__ __forceinline__ void wait_async_0() {
    asm volatile("s_wait_asynccnt 0" ::: "memory");
}

#if HAVE_TDM
// ---- TDM staging (Tensor Data Mover, TENSORcnt) ---------------------------
// Each wave issues ONE tensor_load_to_lds for a RPW-row x COLS-col slab.
// D# layout per CDNA5 ISA.
__device__ __forceinline__ void tdm_load_slab(const float* gptr, unsigned ldsaddr, int B) {
    const unsigned long long ga = (unsigned long long)(const void*)gptr;
    v4u g0;
    g0[0] = 1u;                                          // count=1 (valid), no gather
    g0[1] = ldsaddr;                                     // LDS byte address
    g0[2] = (unsigned)(ga & 0xFFFFFFFFu);                // global_addr[31:0]
    g0[3] = (unsigned)((ga >> 32) & 0x01FFFFFFu)         // global_addr[56:32]
          | 0x80000000u;                                 // type=2 ("image")
    const unsigned TD0 = (unsigned)B;                    // tensor_dim0 (elements)
    const unsigned TD1 = 1u << 22;                       // tensor_dim1 (ample, no OOB)
    v8i g1;
    g1[0] = 0x00020000;                                  // data_size=2 (4B); mask=0
    g1[1] = (int)((TD0 & 0xFFFFu) << 16);                // tensor_dim0[15:0]
    g1[2] = (int)((TD0 >> 16) | ((TD1 & 0xFFFFu) << 16));// dim0[31:16] | dim1[15:0]
    g1[3] = (int)((TD1 >> 16) | ((unsigned)COLS << 16)); // dim1[31:16] | tile_dim0
    g1[4] = RPW;                                         // tile_dim1=RPW, tile_dim2=0
    g1[5] = B;                                           // tensor_dim0_stride[31:0]
    g1[6] = 0;                                           // stride0[47:32] | stride1[15:0]
    g1[7] = 0;
    v4i g2 = (v4i){1, 1, 0, 0};                          // tensor_dim2=1, tensor_dim3=1
    v4i g3 = (v4i){0, 0x00010000, 0, 0};                 // tensor_dim4=1
#if defined(__clang_major__) && (__clang_major__ >= 23)
    __builtin_amdgcn_tensor_load_to_lds(g0, g1, g2, g3, (v8i){0,0,0,0,0,0,0,0}, 0);
#else
    __builtin_amdgcn_tensor_load_to_lds(g0, g1, g2, g3, 0);
#endif
}
#endif  // HAVE_TDM

// ---------------- Phase 1: per-(chunk, column) affine composition ----------------
__global__ void __launch_bounds__(COLS)
delta_phase1(const float* __restrict__ x, const float* __restrict__ lrp,
             float* __restrict__ A0, float* __restrict__ B0,
             float* __restrict__ A1, float* __restrict__ B1, int B) {
    __shared__ float buf[2][STEP][COLS];
    const int tid  = threadIdx.x;
    const int c    = blockIdx.y;
    const int col0 = blockIdx.x * COLS;
    const int b    = col0 + tid;
    const int t0   = c * CHUNK;

    const float lr = clamp01(*lrp);
    float op = (t0 > 0) ? x[(size_t)(t0 - 1) * B + b] : 0.0f;

    float a0 = 1.0f, bb0 = 0.0f, a1 = 1.0f, bb1 = 0.0f;

    const unsigned lb0 = lds_off(&buf[0][0][0]);
    const unsigned lb1 = lds_off(&buf[1][0][0]);

    issue_tile_async(x, t0, B, col0, lb0, tid);
    for (int tile = 0; tile < NTILES; ++tile) {
        if (tile + 1 < NTILES) {
            issue_tile_async(x, t0 + (tile + 1) * STEP, B, col0,
                             ((tile + 1) & 1) ? lb1 : lb0, tid);
            wait_async_le4();
        } else {
            wait_async_0();
        }
        __syncthreads();
        const float* bp = &buf[tile & 1][0][tid];
#pragma unroll
        for (int r = 0; r < STEP; ++r) {
            float xt = bp[r * COLS];
            float g0 = lr * (1.0f - op);
            float g1 = lr * op;
            float m0 = 1.0f - g0;
            float m1 = 1.0f - g1;
            a0 *= m0;  bb0 = fmaf(bb0, m0, g0 * xt);
            a1 *= m1;  bb1 = fmaf(bb1, m1, g1 * xt);
            op = xt;
        }
        __syncthreads();
    }
    size_t idx = (size_t)c * B + b;
    A0[idx] = a0; B0[idx] = bb0; A1[idx] = a1; B1[idx] = bb1;
}

// ---------------- Phase 2: serial composition over chunks (tiny) ----------------
__global__ void delta_phase2(const float* __restrict__ A0, const float* __restrict__ B0,
                             const float* __restrict__ A1, const float* __restrict__ B1,
                             float* __restrict__ S0, float* __restrict__ S1,
                             int B, int NC) {
    int b = blockIdx.x * blockDim.x + threadIdx.x;
    if (b >= B) return;
    float p0 = 0.5f, p1 = 0.5f;
    for (int c = 0; c < NC; ++c) {
        size_t i = (size_t)c * B + b;
        S0[i] = p0;
        S1[i] = p1;
        p0 = fmaf(A0[i], p0, B0[i]);
        p1 = fmaf(A1[i], p1, B1[i]);
    }
}

// ---------------- Phase 3: replay each chunk from its entry state ----------------
__global__ void __launch_bounds__(COLS)
delta_phase3(const float* __restrict__ x, const float* __restrict__ lrp,
             const float* __restrict__ S0, const float* __restrict__ S1,
             float* __restrict__ out, int B) {
    __shared__ float buf[2][STEP][COLS];
    const int tid  = threadIdx.x;
    const int c    = blockIdx.y;
    const int col0 = blockIdx.x * COLS;
    const int b    = col0 + tid;
    const int t0   = c * CHUNK;

    const float lr = clamp01(*lrp);
    float p0 = S0[(size_t)c * B + b];
    float p1 = S1[(size_t)c * B + b];
    float op = (t0 > 0) ? x[(size_t)(t0 - 1) * B + b] : 0.0f;

    const unsigned lb0 = lds_off(&buf[0][0][0]);
    const unsigned lb1 = lds_off(&buf[1][0][0]);

#if HAVE_TDM
    // Per-wave slab: wave w stages rows [w*RPW, w*RPW+RPW) of each tile via TDM.
    const unsigned wid = (unsigned)__builtin_amdgcn_readfirstlane((int)(threadIdx.x >> 5));
    const unsigned lslab = wid * (unsigned)(RPW * COLS * 4);
    tdm_load_slab(x + (size_t)(t0 + wid * RPW) * B + col0, lb0 + lslab, B);
    for (int tile = 0; tile < NTILES; ++tile) {
        if (tile + 1 < NTILES) {
            tdm_load_slab(x + (size_t)(t0 + (tile + 1) * STEP + wid * RPW) * B + col0,
                          (((tile + 1) & 1) ? lb1 : lb0) + lslab, B);
            __builtin_amdgcn_s_wait_tensorcnt(1);   // this wave's previous slab done
        } else {
            __builtin_amdgcn_s_wait_tensorcnt(0);
        }
        asm volatile("" ::: "memory");              // TDM wrote LDS behind clang's back
        __syncthreads();
#else
    issue_tile_async(x, t0, B, col0, lb0, tid);
    for (int tile = 0; tile < NTILES; ++tile) {
        if (tile + 1 < NTILES) {
            issue_tile_async(x, t0 + (tile + 1) * STEP, B, col0,
                             ((tile + 1) & 1) ? lb1 : lb0, tid);
            wait_async_le4();
        } else {
            wait_async_0();
        }
        __syncthreads();
#endif
        const float* bp = &buf[tile & 1][0][tid];
        float* orow = out + (size_t)(t0 + tile * STEP) * B + b;
#pragma unroll
        for (int r = 0; r < STEP; ++r) {
            float xt = bp[r * COLS];
            float g0 = lr * (1.0f - op);
            float g1 = lr * op;
            p0 = fmaf(g0, xt - p0, p0);
            p1 = fmaf(g1, xt - p1, p1);
            float pred = p0 * (1.0f - xt) + p1 * xt;
            op = xt;
            // Non-temporal: preds never re-read; keep x resident in L2.
            __builtin_nontemporal_store(pred, &orow[(size_t)r * B]);
        }
        __syncthreads();
    }
}

// ---------------- Fallback: one thread per column, serial over T ----------------
__global__ void delta_serial(const float* __restrict__ x, const float* __restrict__ lrp,
                             float* __restrict__ out, int T, int B) {
    int b = blockIdx.x * blockDim.x + threadIdx.x;
    if (b >= B) return;
    const float lr = clamp01(*lrp);
    float p0 = 0.5f, p1 = 0.5f, op = 0.0f;
    for (int t = 0; t < T; ++t) {
        float xt = x[(size_t)t * B + b];
        float g0 = lr * (1.0f - op);
        float g1 = lr * op;
        p0 = fmaf(g0, xt - p0, p0);
        p1 = fmaf(g1, xt - p1, p1);
        out[(size_t)t * B + b] = p0 * (1.0f - xt) + p1 * xt;
        op = xt;
    }
}

extern "C" void kernel_launch(void* const* d_in, const int* in_sizes, int n_in,
                              void* d_out, int out_size, void* d_ws, size_t ws_size,
                              hipStream_t stream) {
    const float* x   = (const float*)d_in[0];
    const float* lrp = (const float*)d_in[1];
    float* out = (float*)d_out;

    const int B = BCOLS;
    const long long total = in_sizes[0];
    const int T = (int)(total / B);
    const int NC = T / CHUNK;

    const size_t plane = (size_t)NC * B;              // floats per plane
    const size_t need  = 6 * plane * sizeof(float);   // A0,B0,A1,B1,S0,S1

    const bool fits = ((long long)T * B == total) && (T % CHUNK == 0) &&
                      (B % COLS == 0) && (ws_size >= need);

    if (fits) {
        float* A0 = (float*)d_ws;
        float* B0 = A0 + plane;
        float* A1 = B0 + plane;
        float* B1 = A1 + plane;
        float* S0 = B1 + plane;
        float* S1 = S0 + plane;

        dim3 blk(COLS);
        dim3 grid(B / COLS, NC);
        delta_phase1<<<grid, blk, 0, stream>>>(x, lrp, A0, B0, A1, B1, B);
        delta_phase2<<<dim3((B + 255) / 256), dim3(256), 0, stream>>>(A0, B0, A1, B1,
                                                                      S0, S1, B, NC);
        delta_phase3<<<grid, blk, 0, stream>>>(x, lrp, S0, S1, out, B);
    } else {
        delta_serial<<<dim3((B + 255) / 256), dim3(256), 0, stream>>>(x, lrp, out, T, B);
    }
}